// PointNet2_ReID_85117661872238
// MI455X (gfx1250) — compile-verified
//
#include <hip/hip_runtime.h>
#include <math.h>

typedef __attribute__((ext_vector_type(2))) float v2f;
typedef __attribute__((ext_vector_type(8))) float v8f;

#define EPS_BN 1e-5f

// ---------------------------------------------------------------------------
// Farthest point sampling. One block (256 thr) per batch. xyz staged in LDS.
// ---------------------------------------------------------------------------
__global__ void fps_kernel(const float* __restrict__ xyz, float* __restrict__ newxyz,
                           int Npts, int Snp) {
  __shared__ float sx[2048], sy[2048], sz[2048];
  __shared__ float rv[256];
  __shared__ int   ri[256];
  __shared__ int   sFar;
  int b = blockIdx.x, t = threadIdx.x;
  const float* base = xyz + (size_t)b * Npts * 3;
  for (int i = t; i < Npts; i += 256) {
    sx[i] = base[i * 3 + 0];
    sy[i] = base[i * 3 + 1];
    sz[i] = base[i * 3 + 2];
  }
  float d[8];
  for (int j = 0; j < 8; ++j) d[j] = (t + j * 256 < Npts) ? 1e10f : -1.0f;
  __syncthreads();
  int far = 0;
  for (int s = 0; s < Snp; ++s) {
    float cx = sx[far], cy = sy[far], cz = sz[far];
    if (t < 3) newxyz[((size_t)b * Snp + s) * 3 + t] = (t == 0) ? cx : ((t == 1) ? cy : cz);
    float best = -2.0f; int bi = 0;
    for (int j = 0; j < 8; ++j) {
      int i = t + j * 256;
      if (i < Npts) {
        float dx = sx[i] - cx, dy = sy[i] - cy, dz = sz[i] - cz;
        float dd = dx * dx + dy * dy + dz * dz;
        d[j] = fminf(d[j], dd);
      }
      if (d[j] > best) { best = d[j]; bi = t + j * 256; }
    }
    rv[t] = best; ri[t] = bi;
    __syncthreads();
    for (int off = 128; off > 0; off >>= 1) {
      if (t < off) {
        if (rv[t + off] > rv[t] || (rv[t + off] == rv[t] && ri[t + off] < ri[t])) {
          rv[t] = rv[t + off]; ri[t] = ri[t + off];
        }
      }
      __syncthreads();
    }
    if (t == 0) sFar = ri[0];
    __syncthreads();
    far = sFar;
  }
}

// ---------------------------------------------------------------------------
// Ball query: one thread per (b,s); sequential scan == reference's sorted
// in-range index semantics.
// ---------------------------------------------------------------------------
__global__ void ball_query_kernel(const float* __restrict__ xyz, const float* __restrict__ newxyz,
                                  int* __restrict__ idx, int B, int Npts, int S, int K, float r2) {
  int q = blockIdx.x * blockDim.x + threadIdx.x;
  if (q >= B * S) return;
  int b = q / S;
  const float* nx = newxyz + (size_t)q * 3;
  float cx = nx[0], cy = nx[1], cz = nx[2];
  const float* base = xyz + (size_t)b * Npts * 3;
  int* out = idx + (size_t)q * K;
  int cnt = 0, first = 0;
  for (int j = 0; j < Npts && cnt < K; ++j) {
    float dx = base[j * 3 + 0] - cx;
    float dy = base[j * 3 + 1] - cy;
    float dz = base[j * 3 + 2] - cz;
    float dd = dx * dx + dy * dy + dz * dz;
    if (dd <= r2) { if (cnt == 0) first = j; out[cnt++] = j; }
  }
  if (cnt == 0) { for (int k = 0; k < K; ++k) out[k] = 0; }
  else          { for (int k = cnt; k < K; ++k) out[k] = first; }
}

// ---------------------------------------------------------------------------
// Grouping kernels. X layout: [row][n], n=(b*S+s)*K+k, row stride B*S*K.
// ---------------------------------------------------------------------------
__global__ void group_xyz_kernel(const float* __restrict__ xyz, const float* __restrict__ newxyz,
                                 const int* __restrict__ idx, float* __restrict__ X,
                                 int B, int Npts, int S, int K) {
  size_t tot = (size_t)B * S * K;
  size_t n = (size_t)blockIdx.x * blockDim.x + threadIdx.x;
  if (n >= tot) return;
  size_t sk = n / K;
  int s = (int)(sk % S), b = (int)(sk / S);
  int i = idx[n];
  const float* p = xyz + ((size_t)b * Npts + i) * 3;
  const float* q = newxyz + ((size_t)b * S + s) * 3;
  X[n]           = p[0] - q[0];
  X[tot + n]     = p[1] - q[1];
  X[2 * tot + n] = p[2] - q[2];
}

__global__ void group_feat_kernel(const float* __restrict__ feat, const int* __restrict__ idx,
                                  float* __restrict__ X, int B, int C, int Npts, int S, int K,
                                  int rowoff) {
  size_t tot = (size_t)B * S * K;
  size_t n = (size_t)blockIdx.x * blockDim.x + threadIdx.x;
  if (n >= tot) return;
  size_t sk = n / K;
  int b = (int)(sk / S);
  int i = idx[n];
  const float* fb = feat + (size_t)b * C * Npts;
  for (int c = 0; c < C; ++c)
    X[(size_t)(rowoff + c) * tot + n] = fb[(size_t)c * Npts + i];
}

__global__ void group_all_kernel(const float* __restrict__ xyz2, const float* __restrict__ feat,
                                 float* __restrict__ X, int B, int C, int K) {
  int tot = B * K;
  int n = blockIdx.x * blockDim.x + threadIdx.x;
  if (n >= tot) return;
  int k = n % K, b = n / K;
  const float* p = xyz2 + ((size_t)b * K + k) * 3;
  X[n] = p[0]; X[tot + n] = p[1]; X[2 * tot + n] = p[2];
  const float* fb = feat + (size_t)b * C * K;
  for (int c = 0; c < C; ++c)
    X[(size_t)(3 + c) * tot + n] = fb[(size_t)c * K + k];
}

// ---------------------------------------------------------------------------
// Prep kernels: zero-fill / pad so the GEMM hot loop is totally branch-free.
// ---------------------------------------------------------------------------
__global__ void zero_kernel(float* __restrict__ p, size_t n) {
  size_t i = (size_t)blockIdx.x * blockDim.x + threadIdx.x;
  if (i < n) p[i] = 0.f;
}

__global__ void pad_weight_kernel(const float* __restrict__ W, float* __restrict__ Wp,
                                  int M, int K, int Mp, int K4) {
  int i = blockIdx.x * blockDim.x + threadIdx.x;
  if (i >= Mp * K4) return;
  int r = i / K4, c = i % K4;
  Wp[i] = (r < M && c < K) ? W[(size_t)r * K + c] : 0.f;
}

__global__ void pad_bias_kernel(const float* __restrict__ b, float* __restrict__ bp,
                                int M, int Mp) {
  int i = blockIdx.x * blockDim.x + threadIdx.x;
  if (i >= Mp) return;
  bp[i] = (i < M) ? b[i] : 0.f;
}

// ---------------------------------------------------------------------------
// fp32 WMMA GEMM + bias, fully guard-free:
//   Y[Mp][N] = W[Mp][K4] * X[K4][N] + bias[Mp]
// Preconditions: Mp % (16*TM) == 0, K4 % 4 == 0, N % (16*TN) == 0; pad rows
// of W/X are zero (prep kernels).
// One wave computes a (16*TM) x (16*TN) tile. Per k-group all fragments are
// loaded into registers first (one load cluster -> one s_wait), then TM*TN
// WMMAs issue back-to-back: at TM=2,TN=4 that is 8 wmma per 10 loads, with
// A reused across TN tiles and B reused across TM tiles.
// Lane layouts:
//   A 16x4 : a[v] = A[lane&15][(lane>>4)*2 + v]
//   B 4x16 : b[v] = B[(lane>>4)*2 + v][lane&15]
//   C/D    : d[r] = D[r + 8*(lane>>4)][lane&15]
// ---------------------------------------------------------------------------
template <int TM, int TN>
__global__ void wmma_gemm_bias_kernel(const float* __restrict__ W, const float* __restrict__ X,
                                      const float* __restrict__ bias, float* __restrict__ Y,
                                      int K4, int N, int tilesM, int tilesN) {
  long gw = ((long)blockIdx.x * blockDim.x + threadIdx.x) >> 5;
  if (gw >= (long)tilesM * tilesN) return;
  int lane = threadIdx.x & 31;
  int tm = (int)(gw / tilesN) * (16 * TM);
  int tn = (int)(gw % tilesN) * (16 * TN);
  int l16 = lane & 15;
  int hi  = lane >> 4;
  const float* Wp = W + (size_t)(tm + l16) * K4 + (hi << 1);  // 8B aligned
  const float* Xp = X + (size_t)(hi << 1) * N + tn + l16;
  v8f acc[TM][TN];
#pragma unroll
  for (int i = 0; i < TM; ++i)
#pragma unroll
    for (int j = 0; j < TN; ++j) acc[i][j] = (v8f){0.f, 0.f, 0.f, 0.f, 0.f, 0.f, 0.f, 0.f};
#pragma unroll 2
  for (int k0 = 0; k0 < K4; k0 += 4) {
    // ---- load cluster: all fragments for this k-group ----
    v2f a[TM];
#pragma unroll
    for (int i = 0; i < TM; ++i) a[i] = *(const v2f*)(Wp + (size_t)i * 16 * K4 + k0);
    const float* xp = Xp + (size_t)k0 * N;
    v2f b[TN];
#pragma unroll
    for (int j = 0; j < TN; ++j) {
      b[j].x = xp[j * 16];
      b[j].y = xp[(size_t)N + j * 16];
    }
    // ---- compute cluster: TM*TN WMMAs back-to-back ----
#pragma unroll
    for (int i = 0; i < TM; ++i)
#pragma unroll
      for (int j = 0; j < TN; ++j)
        acc[i][j] = __builtin_amdgcn_wmma_f32_16x16x4_f32(false, a[i], false, b[j], (short)0,
                                                          acc[i][j], false, false);
  }
#pragma unroll
  for (int i = 0; i < TM; ++i) {
#pragma unroll
    for (int r = 0; r < 8; ++r) {
      int row = tm + i * 16 + r + (hi << 3);
      float bb = bias[row];
      float* yp = Y + (size_t)row * N + tn + l16;
#pragma unroll
      for (int j = 0; j < TN; ++j) yp[j * 16] = acc[i][j][r] + bb;
    }
  }
}

// ---------------------------------------------------------------------------
// BatchNorm (batch statistics) + ReLU over [M][Ncols], per row.
// ---------------------------------------------------------------------------
__global__ void bn_stats_kernel(const float* __restrict__ Y, float* __restrict__ mean,
                                float* __restrict__ var, int Ncols) {
  __shared__ float ss[256], sq[256];
  int row = blockIdx.x, t = threadIdx.x;
  const float* p = Y + (size_t)row * Ncols;
  float s = 0.f, q = 0.f;
  for (int i = t; i < Ncols; i += 256) { float v = p[i]; s += v; q += v * v; }
  ss[t] = s; sq[t] = q;
  __syncthreads();
  for (int off = 128; off > 0; off >>= 1) {
    if (t < off) { ss[t] += ss[t + off]; sq[t] += sq[t + off]; }
    __syncthreads();
  }
  if (t == 0) {
    float m = ss[0] / (float)Ncols;
    mean[row] = m;
    var[row] = fmaxf(sq[0] / (float)Ncols - m * m, 0.f);
  }
}

__global__ void bn_apply_relu_kernel(float* __restrict__ Y, const float* __restrict__ mean,
                                     const float* __restrict__ var, const float* __restrict__ g,
                                     const float* __restrict__ bt, int M, int Ncols) {
  size_t tot = (size_t)M * Ncols;
  size_t i = (size_t)blockIdx.x * blockDim.x + threadIdx.x;
  if (i >= tot) return;
  int row = (int)(i / (size_t)Ncols);
  float v = Y[i];
  v = g[row] * (v - mean[row]) * rsqrtf(var[row] + EPS_BN) + bt[row];
  Y[i] = fmaxf(v, 0.f);
}

// ---------------------------------------------------------------------------
// Max pool over k: Y[C][B*S*K] -> out[b][c][s]
// ---------------------------------------------------------------------------
__global__ void maxpool_kernel(const float* __restrict__ Y, float* __restrict__ out,
                               int B, int C, int S, int K) {
  int i = blockIdx.x * blockDim.x + threadIdx.x;
  if (i >= B * C * S) return;
  int s = i % S, c = (i / S) % C, b = i / (S * C);
  const float* p = Y + (size_t)c * B * S * K + ((size_t)b * S + s) * (size_t)K;
  float m = p[0];
  for (int k = 1; k < K; ++k) m = fmaxf(m, p[k]);
  out[i] = m;
}

__global__ void maxpool_T_kernel(const float* __restrict__ Y, float* __restrict__ out,
                                 int C, int B, int K) {
  int i = blockIdx.x * blockDim.x + threadIdx.x;
  if (i >= C * B) return;
  int b = i % B, c = i / B;
  const float* p = Y + (size_t)c * B * K + (size_t)b * K;
  float m = p[0];
  for (int k = 1; k < K; ++k) m = fmaxf(m, p[k]);
  out[i] = m;
}

// ---------------------------------------------------------------------------
// log_softmax over rows (classes) per batch column; logits layout [M][B]
// ---------------------------------------------------------------------------
__global__ void log_softmax_col_kernel(const float* __restrict__ Yl, float* __restrict__ out,
                                       int M, int B) {
  __shared__ float red[256];
  int b = blockIdx.x, t = threadIdx.x;
  float mx = -1e30f;
  for (int f = t; f < M; f += 256) mx = fmaxf(mx, Yl[(size_t)f * B + b]);
  red[t] = mx;
  __syncthreads();
  for (int off = 128; off > 0; off >>= 1) {
    if (t < off) red[t] = fmaxf(red[t], red[t + off]);
    __syncthreads();
  }
  mx = red[0];
  __syncthreads();
  float s = 0.f;
  for (int f = t; f < M; f += 256) s += expf(Yl[(size_t)f * B + b] - mx);
  red[t] = s;
  __syncthreads();
  for (int off = 128; off > 0; off >>= 1) {
    if (t < off) red[t] += red[t + off];
    __syncthreads();
  }
  float lse = mx + logf(red[0]);
  for (int f = t; f < M; f += 256) out[(size_t)b * M + f] = Yl[(size_t)f * B + b] - lse;
}

__global__ void copy_T_kernel(const float* __restrict__ Y, float* __restrict__ out, int M, int B) {
  int i = blockIdx.x * blockDim.x + threadIdx.x;
  if (i >= M * B) return;
  int b = i / M, f = i % M;
  out[i] = Y[(size_t)f * B + b];
}

// ---------------------------------------------------------------------------
static inline void run_gemm(const float* W, const float* X, const float* bias, float* Y,
                            int Mp, int K4, int N, hipStream_t s) {
  int tilesM = Mp / 32;  // Mp always a multiple of 32 here
  if (N % 64 == 0) {
    int tilesN = N / 64;
    long tiles = (long)tilesM * tilesN;
    wmma_gemm_bias_kernel<2, 4><<<(unsigned)((tiles + 7) / 8), 256, 0, s>>>(W, X, bias, Y, K4, N,
                                                                            tilesM, tilesN);
  } else {  // head layers, N == 32
    int tilesN = N / 32;
    long tiles = (long)tilesM * tilesN;
    wmma_gemm_bias_kernel<2, 2><<<(unsigned)((tiles + 7) / 8), 256, 0, s>>>(W, X, bias, Y, K4, N,
                                                                            tilesM, tilesN);
  }
}

static inline void run_bn_relu(float* Y, const float* g, const float* bt, int M, int Ncols,
                               float* meanB, float* varB, hipStream_t s) {
  bn_stats_kernel<<<M, 256, 0, s>>>(Y, meanB, varB, Ncols);
  size_t tot = (size_t)M * Ncols;
  bn_apply_relu_kernel<<<(unsigned)((tot + 255) / 256), 256, 0, s>>>(Y, meanB, varB, g, bt, M,
                                                                     Ncols);
}

extern "C" void kernel_launch(void* const* d_in, const int* in_sizes, int n_in,
                              void* d_out, int out_size, void* d_ws, size_t ws_size,
                              hipStream_t stream) {
  (void)in_sizes; (void)n_in; (void)out_size; (void)ws_size;
  const float* pts = (const float*)d_in[0];
  const float* sa1_w[3]  = {(const float*)d_in[1],  (const float*)d_in[2],  (const float*)d_in[3]};
  const float* sa1_b[3]  = {(const float*)d_in[4],  (const float*)d_in[5],  (const float*)d_in[6]};
  const float* sa1_g[3]  = {(const float*)d_in[7],  (const float*)d_in[8],  (const float*)d_in[9]};
  const float* sa1_bt[3] = {(const float*)d_in[10], (const float*)d_in[11], (const float*)d_in[12]};
  const float* sa2_w[3]  = {(const float*)d_in[13], (const float*)d_in[14], (const float*)d_in[15]};
  const float* sa2_b[3]  = {(const float*)d_in[16], (const float*)d_in[17], (const float*)d_in[18]};
  const float* sa2_g[3]  = {(const float*)d_in[19], (const float*)d_in[20], (const float*)d_in[21]};
  const float* sa2_bt[3] = {(const float*)d_in[22], (const float*)d_in[23], (const float*)d_in[24]};
  const float* sa3_w[3]  = {(const float*)d_in[25], (const float*)d_in[26], (const float*)d_in[27]};
  const float* sa3_b[3]  = {(const float*)d_in[28], (const float*)d_in[29], (const float*)d_in[30]};
  const float* sa3_g[3]  = {(const float*)d_in[31], (const float*)d_in[32], (const float*)d_in[33]};
  const float* sa3_bt[3] = {(const float*)d_in[34], (const float*)d_in[35], (const float*)d_in[36]};
  const float* head_w[3] = {(const float*)d_in[37], (const float*)d_in[38], (const float*)d_in[39]};
  const float* head_b[3] = {(const float*)d_in[40], (const float*)d_in[41], (const float*)d_in[42]};
  const float* head_g[2] = {(const float*)d_in[43], (const float*)d_in[44]};
  const float* head_bt[2]= {(const float*)d_in[45], (const float*)d_in[46]};

  const int B = 32, N0 = 2048, S1 = 512, K1 = 32, S2 = 128, K2 = 64;
  const int N1 = B * S1 * K1;       // 524288
  const int N2 = B * S2 * K2;       // 262144
  const int N3 = B * 128;           // 4096
  const int CLS = 751, CLSp = 768;  // pad classes to multiple of 32 (16*TM)

  // ---- workspace layout (floats), ~690 MB ----
  float* wsf = (float*)d_ws;
  size_t off = 0;
  float* Xbuf     = wsf + off; off += 34603008;  // max(4*N1, 132*N2, 260*N3)
  float* Ybuf1    = wsf + off; off += 33554432;  // max(64*N1, 128*N2, ...)
  float* Ybuf2    = wsf + off; off += 33554432;
  float* Ybuf3    = wsf + off; off += 67108864;  // max(128*N1, 256*N2, 1024*N3)
  float* pooled1  = wsf + off; off += (size_t)B * 128 * S1;   // [b][c][s]
  float* pooled2  = wsf + off; off += (size_t)B * 256 * S2;
  float* pooled3T = wsf + off; off += (size_t)1024 * B;        // [c][b]
  float* newxyz1  = wsf + off; off += (size_t)B * S1 * 3;
  float* newxyz2  = wsf + off; off += (size_t)B * S2 * 3;
  float* meanB    = wsf + off; off += 1024;
  float* varB     = wsf + off; off += 1024;
  float* w1p      = wsf + off; off += 64 * 4;        // sa1_w[0] padded K 3->4
  float* w2p      = wsf + off; off += 128 * 132;     // sa2_w[0] padded K 131->132
  float* w3p      = wsf + off; off += 256 * 260;     // sa3_w[0] padded K 259->260
  float* hw2p     = wsf + off; off += (size_t)CLSp * 256;  // head_w[2] padded M 751->768
  float* hb2p     = wsf + off; off += CLSp;
  int* idx1 = (int*)(wsf + off); off += (size_t)B * S1 * K1;
  int* idx2 = (int*)(wsf + off); off += (size_t)B * S2 * K2;

  // ---- pad weights/bias once (tiny) ----
  pad_weight_kernel<<<(64 * 4 + 255) / 256, 256, 0, stream>>>(sa1_w[0], w1p, 64, 3, 64, 4);
  pad_weight_kernel<<<(128 * 132 + 255) / 256, 256, 0, stream>>>(sa2_w[0], w2p, 128, 131, 128, 132);
  pad_weight_kernel<<<(256 * 260 + 255) / 256, 256, 0, stream>>>(sa3_w[0], w3p, 256, 259, 256, 260);
  pad_weight_kernel<<<(CLSp * 256 + 255) / 256, 256, 0, stream>>>(head_w[2], hw2p, CLS, 256, CLSp, 256);
  pad_bias_kernel<<<(CLSp + 255) / 256, 256, 0, stream>>>(head_b[2], hb2p, CLS, CLSp);

  // ================= SA1 =================
  fps_kernel<<<B, 256, 0, stream>>>(pts, newxyz1, N0, S1);
  ball_query_kernel<<<(B * S1 + 255) / 256, 256, 0, stream>>>(pts, newxyz1, idx1, B, N0, S1, K1, 0.04f);
  group_xyz_kernel<<<(N1 + 255) / 256, 256, 0, stream>>>(pts, newxyz1, idx1, Xbuf, B, N0, S1, K1);
  zero_kernel<<<(N1 + 255) / 256, 256, 0, stream>>>(Xbuf + (size_t)3 * N1, (size_t)N1);  // pad row
  run_gemm(w1p, Xbuf, sa1_b[0], Ybuf1, 64, 4, N1, stream);
  run_bn_relu(Ybuf1, sa1_g[0], sa1_bt[0], 64, N1, meanB, varB, stream);
  run_gemm(sa1_w[1], Ybuf1, sa1_b[1], Ybuf2, 64, 64, N1, stream);
  run_bn_relu(Ybuf2, sa1_g[1], sa1_bt[1], 64, N1, meanB, varB, stream);
  run_gemm(sa1_w[2], Ybuf2, sa1_b[2], Ybuf3, 128, 64, N1, stream);
  run_bn_relu(Ybuf3, sa1_g[2], sa1_bt[2], 128, N1, meanB, varB, stream);
  maxpool_kernel<<<(B * 128 * S1 + 255) / 256, 256, 0, stream>>>(Ybuf3, pooled1, B, 128, S1, K1);

  // ================= SA2 =================
  fps_kernel<<<B, 256, 0, stream>>>(newxyz1, newxyz2, S1, S2);
  ball_query_kernel<<<(B * S2 + 255) / 256, 256, 0, stream>>>(newxyz1, newxyz2, idx2, B, S1, S2, K2, 0.16f);
  group_xyz_kernel<<<(N2 + 255) / 256, 256, 0, stream>>>(newxyz1, newxyz2, idx2, Xbuf, B, S1, S2, K2);
  group_feat_kernel<<<(N2 + 255) / 256, 256, 0, stream>>>(pooled1, idx2, Xbuf, B, 128, S1, S2, K2, 3);
  zero_kernel<<<(N2 + 255) / 256, 256, 0, stream>>>(Xbuf + (size_t)131 * N2, (size_t)N2);  // pad row
  run_gemm(w2p, Xbuf, sa2_b[0], Ybuf1, 128, 132, N2, stream);
  run_bn_relu(Ybuf1, sa2_g[0], sa2_bt[0], 128, N2, meanB, varB, stream);
  run_gemm(sa2_w[1], Ybuf1, sa2_b[1], Ybuf2, 128, 128, N2, stream);
  run_bn_relu(Ybuf2, sa2_g[1], sa2_bt[1], 128, N2, meanB, varB, stream);
  run_gemm(sa2_w[2], Ybuf2, sa2_b[2], Ybuf3, 256, 128, N2, stream);
  run_bn_relu(Ybuf3, sa2_g[2], sa2_bt[2], 256, N2, meanB, varB, stream);
  maxpool_kernel<<<(B * 256 * S2 + 255) / 256, 256, 0, stream>>>(Ybuf3, pooled2, B, 256, S2, K2);

  // ================= SA3 (group_all) =================
  group_all_kernel<<<(N3 + 255) / 256, 256, 0, stream>>>(newxyz2, pooled2, Xbuf, B, 256, 128);
  zero_kernel<<<(N3 + 255) / 256, 256, 0, stream>>>(Xbuf + (size_t)259 * N3, (size_t)N3);  // pad row
  run_gemm(w3p, Xbuf, sa3_b[0], Ybuf1, 256, 260, N3, stream);
  run_bn_relu(Ybuf1, sa3_g[0], sa3_bt[0], 256, N3, meanB, varB, stream);
  run_gemm(sa3_w[1], Ybuf1, sa3_b[1], Ybuf2, 512, 256, N3, stream);
  run_bn_relu(Ybuf2, sa3_g[1], sa3_bt[1], 512, N3, meanB, varB, stream);
  run_gemm(sa3_w[2], Ybuf2, sa3_b[2], Ybuf3, 1024, 512, N3, stream);
  run_bn_relu(Ybuf3, sa3_g[2], sa3_bt[2], 1024, N3, meanB, varB, stream);
  maxpool_T_kernel<<<(1024 * B + 255) / 256, 256, 0, stream>>>(Ybuf3, pooled3T, 1024, B, 128);

  // ================= Head =================
  float* outp = (float*)d_out;
  run_gemm(head_w[0], pooled3T, head_b[0], Ybuf1, 256, 1024, B, stream);
  run_bn_relu(Ybuf1, head_g[0], head_bt[0], 256, B, meanB, varB, stream);
  run_gemm(head_w[1], Ybuf1, head_b[1], Ybuf2, 256, 256, B, stream);
  run_bn_relu(Ybuf2, head_g[1], head_bt[1], 256, B, meanB, varB, stream);
  copy_T_kernel<<<(256 * B + 255) / 256, 256, 0, stream>>>(Ybuf2, outp + (size_t)B * CLS, 256, B);
  run_gemm(hw2p, Ybuf2, hb2p, Ybuf3, CLSp, 256, B, stream);  // pad rows: W,b zero -> unread
  log_softmax_col_kernel<<<B, 256, 0, stream>>>(Ybuf3, outp, CLS, B);
}